// EdgePooling_59107339927792
// MI455X (gfx1250) — compile-verified
//
#include <hip/hip_runtime.h>
#include <hip/hip_bf16.h>

typedef float v2f __attribute__((ext_vector_type(2)));
typedef float v8f __attribute__((ext_vector_type(8)));

#define RADIX_ITEMS 4096          // 256 threads * 16 items
#define MAXN_WORDS 2048           // supports N up to 65536 (N = 50000)
#define WLDS_FLOATS 2048          // >= 3*C (C = 256 -> 768 used)

// ---------------------------------------------------------------------------
// Order-preserving float<->uint transforms (for atomic max + radix keys)
// ---------------------------------------------------------------------------
__device__ __forceinline__ unsigned orderKey(float f) {
    unsigned u = __float_as_uint(f);
    return (u & 0x80000000u) ? ~u : (u | 0x80000000u);   // ascending order
}
__device__ __forceinline__ float fromOrderKey(unsigned v) {
    unsigned u = (v & 0x80000000u) ? (v ^ 0x80000000u) : ~v;
    return __uint_as_float(u);
}

// ---------------------------------------------------------------------------
// s_src = x @ W[:C], s_dst = x @ W[C:], via V_WMMA_F32_16X16X4_F32.
// One wave handles 16 rows. B-matrix columns 0/1 carry the two weight vectors,
// other columns read a zeroed LDS pad; W is staged in LDS once so the inner
// loop is branch-free: ds_load_b64 (B) + global_load_b64 (A) + v_wmma.
// A 16x4 f32 layout: lane L holds row M=L%16, K = (L/16)*2 + {0,1} in 2 VGPRs.
// B 4x16 f32 layout (mirror): lane L holds col N=L%16, K = (L/16)*2 + {0,1}.
// D 16x16 f32: vgpr j, lanes 0-15 -> M=j, lanes 16-31 -> M=j+8; N = lane%16.
// ---------------------------------------------------------------------------
__global__ __launch_bounds__(32) void matvec_wmma_kernel(
    const float* __restrict__ x, const float* __restrict__ W,
    float* __restrict__ s_src, float* __restrict__ s_dst, int N, int C)
{
    __shared__ float wlds[WLDS_FLOATS];   // [0,2C): W ; [2C,3C): zeros
    const int tile = blockIdx.x;
    const int lane = threadIdx.x;
    const int m    = lane & 15;
    const int kb   = (lane >> 4) << 1;            // 0 or 2
    int row  = tile * 16 + m;
    int rowc = row < N ? row : (N - 1);           // clamp loads; stores guarded
    const float* xr = x + (size_t)rowc * C;
    const int n = m;                              // B-matrix column this lane feeds

    for (int i = lane; i < 2 * C; i += 32) wlds[i] = W[i];
    for (int i = 2 * C + lane; i < 3 * C; i += 32) wlds[i] = 0.0f;
    __syncthreads();

    // lane-constant LDS base: col 0 -> W[:C], col 1 -> W[C:2C], cols >=2 -> zeros
    const float* wp = &wlds[(n < 2) ? n * C : 2 * C];

    v8f acc = {};
    for (int k0 = 0; k0 < C; k0 += 4) {
        v2f a;
        a.x = xr[k0 + kb];
        a.y = xr[k0 + kb + 1];
        v2f bm;
        bm.x = wp[k0 + kb];
        bm.y = wp[k0 + kb + 1];
        acc = __builtin_amdgcn_wmma_f32_16x16x4_f32(
            false, a, false, bm, (short)0, acc, false, false);
    }

    // Columns 0 (s_src) and 1 (s_dst) live in lanes {0,16} and {1,17}.
    if (lane == 0 || lane == 1 || lane == 16 || lane == 17) {
        float* outp = (lane & 1) ? s_dst : s_src;
        int mb = tile * 16 + ((lane >> 4) << 3);  // +0 or +8
#pragma unroll
        for (int j = 0; j < 8; ++j) {
            int r = mb + j;
            if (r < N) outp[r] = acc[j];
        }
    }
}

// ---------------------------------------------------------------------------
// Elementwise / init kernels
// ---------------------------------------------------------------------------
__global__ void zero_f32_kernel(float* __restrict__ p, long long count) {
    long long i = (long long)blockIdx.x * blockDim.x + threadIdx.x;
    long long stride = (long long)gridDim.x * blockDim.x;
    for (; i < count; i += stride) p[i] = 0.0f;
}

__global__ void init_nodes_kernel(unsigned* __restrict__ segmax,
                                  unsigned long long* __restrict__ segsum,
                                  float* __restrict__ cl_score, int N) {
    int i = blockIdx.x * blockDim.x + threadIdx.x;
    if (i < N) { segmax[i] = 0u; segsum[i] = 0ull; cl_score[i] = 1.0f; }
}

// raw[e] = s_src[src] + s_dst[dst] + b ; segment max by dst (order-invariant)
__global__ void raw_max_kernel(const int* __restrict__ src, const int* __restrict__ dst,
                               const float* __restrict__ s_src, const float* __restrict__ s_dst,
                               const float* __restrict__ bias,
                               float* __restrict__ raw, unsigned* __restrict__ segmax, int E)
{
    int e = blockIdx.x * blockDim.x + threadIdx.x;
    if (e >= E) return;
    float r = s_src[src[e]] + s_dst[dst[e]] + bias[0];
    raw[e] = r;
    atomicMax(&segmax[dst[e]], orderKey(r));
}

// ex = exp(raw - max); deterministic fixed-point segment sum (x 2^32)
__global__ void exp_sum_kernel(const int* __restrict__ dst,
                               float* __restrict__ raw_ex,
                               const unsigned* __restrict__ segmax,
                               unsigned long long* __restrict__ segsum, int E)
{
    int e = blockIdx.x * blockDim.x + threadIdx.x;
    if (e >= E) return;
    int d = dst[e];
    float m  = fromOrderKey(segmax[d]);
    float ex = __expf(raw_ex[e] - m);
    raw_ex[e] = ex;
    atomicAdd(&segsum[d], (unsigned long long)((double)ex * 4294967296.0));
}

// score = ex / sum + 0.5 ; emit radix keys (descending, stable == argsort(-e))
__global__ void finalize_keys_kernel(const int* __restrict__ dst,
                                     float* __restrict__ ex_score,
                                     const unsigned long long* __restrict__ segsum,
                                     unsigned* __restrict__ keys, unsigned* __restrict__ vals, int E)
{
    int e = blockIdx.x * blockDim.x + threadIdx.x;
    if (e >= E) return;
    double s = (double)segsum[dst[e]] * (1.0 / 4294967296.0);
    float p = (float)((double)ex_score[e] / s) + 0.5f;
    ex_score[e] = p;
    keys[e] = ~orderKey(p);      // ascending radix on this == descending score
    vals[e] = (unsigned)e;
}

// ---------------------------------------------------------------------------
// Stable LSD radix sort (8-bit digits, 4 passes), digit-major histogram
// ---------------------------------------------------------------------------
__global__ __launch_bounds__(256) void radix_hist_kernel(
    const unsigned* __restrict__ keys, unsigned* __restrict__ hist,
    int nItems, int shift, int nblk)
{
    __shared__ unsigned cnt[256];
    int t = threadIdx.x, b = blockIdx.x;
    cnt[t] = 0;
    __syncthreads();
    int base = b * RADIX_ITEMS;
#pragma unroll
    for (int j = 0; j < 16; ++j) {
        int i = base + j * 256 + t;
        if (i < nItems) atomicAdd(&cnt[(keys[i] >> shift) & 255u], 1u);
    }
    __syncthreads();
    hist[t * nblk + b] = cnt[t];
}

__global__ void radix_scan_kernel(unsigned* __restrict__ hist, int total) {
    if (blockIdx.x == 0 && threadIdx.x == 0) {
        unsigned run = 0;
        for (int i = 0; i < total; ++i) { unsigned v = hist[i]; hist[i] = run; run += v; }
    }
}

__global__ __launch_bounds__(256) void radix_scatter_kernel(
    const unsigned* __restrict__ keysIn, const unsigned* __restrict__ valsIn,
    unsigned* __restrict__ keysOut, unsigned* __restrict__ valsOut,
    const unsigned* __restrict__ hist, int nItems, int shift, int nblk)
{
    __shared__ unsigned k[RADIX_ITEMS];
    __shared__ unsigned v[RADIX_ITEMS];
    __shared__ unsigned pos[RADIX_ITEMS];
    __shared__ unsigned lofs[256];
    int t = threadIdx.x, b = blockIdx.x;
    int base = b * RADIX_ITEMS;
    int lim = nItems - base; if (lim > RADIX_ITEMS) lim = RADIX_ITEMS;
#pragma unroll
    for (int j = 0; j < 16; ++j) {
        int idx = j * 256 + t;
        if (idx < lim) { k[idx] = keysIn[base + idx]; v[idx] = valsIn[base + idx]; }
    }
    lofs[t] = hist[t * nblk + b];
    __syncthreads();
    if (t == 0) {                          // stable in-block placement
        for (int j = 0; j < lim; ++j) {
            unsigned d = (k[j] >> shift) & 255u;
            pos[j] = lofs[d]++;
        }
    }
    __syncthreads();
#pragma unroll
    for (int j = 0; j < 16; ++j) {
        int idx = j * 256 + t;
        if (idx < lim) { unsigned p = pos[idx]; keysOut[p] = k[idx]; valsOut[p] = v[idx]; }
    }
}

// ---------------------------------------------------------------------------
// Sequential greedy matching: 50000-bit availability mask in LDS; the 8-wave block
// prefetches 256 sorted edges/chunk into LDS, lane 0 runs the automaton.
// ---------------------------------------------------------------------------
__global__ __launch_bounds__(256) void greedy_kernel(
    const unsigned* __restrict__ vals_sorted,
    const int* __restrict__ src, const int* __restrict__ dst,
    const float* __restrict__ score,
    int* __restrict__ cluster_out, float* __restrict__ cl_score, int N, int E)
{
    __shared__ unsigned mask[MAXN_WORDS];
    __shared__ int sS[256], sD[256];
    __shared__ float sC[256];
    __shared__ int s_cnt;
    int t = threadIdx.x;
    int nw = (N + 31) >> 5;
    for (int i = t; i < nw; i += 256) mask[i] = 0xFFFFFFFFu;
    if (t == 0) s_cnt = 0;
    __syncthreads();

    int chunks = (E + 255) / 256;
    for (int c = 0; c < chunks; ++c) {
        int i = c * 256 + t;
        if (i < E) {
            int vv = (int)vals_sorted[i];
            sS[t] = src[vv]; sD[t] = dst[vv]; sC[t] = score[vv];
        }
        __syncthreads();
        if (t == 0) {
            int cnt = s_cnt;
            int lim = E - c * 256; if (lim > 256) lim = 256;
            for (int j = 0; j < lim; ++j) {
                int s = sS[j], d = sD[j];
                unsigned bs = (mask[s >> 5] >> (s & 31)) & 1u;
                unsigned bd = (mask[d >> 5] >> (d & 31)) & 1u;
                if (bs & bd) {
                    mask[s >> 5] &= ~(1u << (s & 31));
                    mask[d >> 5] &= ~(1u << (d & 31));
                    cluster_out[s] = cnt;
                    cluster_out[d] = cnt;
                    cl_score[cnt] = sC[j];
                    ++cnt;
                }
            }
            s_cnt = cnt;
        }
        __syncthreads();
    }
    if (t == 0) {                      // singleton clusters for survivors
        int cnt = s_cnt;
        for (int n2 = 0; n2 < N; ++n2)
            if ((mask[n2 >> 5] >> (n2 & 31)) & 1u)
                cluster_out[n2] = cnt++;
    }
}

// ---------------------------------------------------------------------------
// Cluster aggregation (clusters have <= 2 members -> atomicAdd deterministic)
// ---------------------------------------------------------------------------
__global__ void newx_scatter_kernel(const float* __restrict__ x,
                                    const int* __restrict__ cluster,
                                    const float* __restrict__ cl_score,
                                    float* __restrict__ newx, int N, int C)
{
    long long i = (long long)blockIdx.x * blockDim.x + threadIdx.x;
    long long total = (long long)N * C;
    if (i >= total) return;
    int n = (int)(i / C), c = (int)(i % C);
    int cl = cluster[n];
    atomicAdd(&newx[(size_t)cl * C + c], x[i] * cl_score[cl]);
}

__global__ void remap_edges_kernel(const int* __restrict__ ei,
                                   const int* __restrict__ cluster,
                                   int* __restrict__ out, int n2E)
{
    int i = blockIdx.x * blockDim.x + threadIdx.x;
    if (i < n2E) out[i] = cluster[ei[i]];
}

__global__ void batch_scatter_kernel(const int* __restrict__ batch,
                                     const int* __restrict__ cluster,
                                     int* __restrict__ newb, int N)
{
    int i = blockIdx.x * blockDim.x + threadIdx.x;
    if (i < N) newb[cluster[i]] = batch[i];
}

// ---------------------------------------------------------------------------
extern "C" void kernel_launch(void* const* d_in, const int* in_sizes, int n_in,
                              void* d_out, int out_size, void* d_ws, size_t ws_size,
                              hipStream_t stream)
{
    const float* x    = (const float*)d_in[0];
    const int*   ei   = (const int*)d_in[1];
    const int*   batch= (const int*)d_in[2];
    const float* W    = (const float*)d_in[3];
    const float* bias = (const float*)d_in[4];

    const int C = in_sizes[3] / 2;          // W is [2C]
    const int N = in_sizes[0] / C;          // x is [N, C]
    const int E = in_sizes[1] / 2;          // edge_index is [2, E]
    const int* src = ei;
    const int* dst = ei + E;

    // ---- output layout (float buffer; int regions reinterpreted) ----
    float* out_f    = (float*)d_out;
    float* newx     = out_f;                              // N*C
    int*   new_ei   = (int*)(out_f + (size_t)N * C);      // 2E
    int*   new_b    = (int*)(out_f + (size_t)N * C + 2 * (size_t)E); // N
    int*   cluster  = (int*)(out_f + (size_t)N * C + 2 * (size_t)E + N); // N

    // ---- workspace layout ----
    char* ws = (char*)d_ws;
    size_t off = 0;
    auto take = [&](size_t bytes) { size_t o = off; off = (off + bytes + 255) & ~(size_t)255; return o; };
    float*              s_src  = (float*)(ws + take((size_t)N * 4));
    float*              s_dst  = (float*)(ws + take((size_t)N * 4));
    float*              score  = (float*)(ws + take((size_t)E * 4));  // raw -> ex -> score
    unsigned*           segmax = (unsigned*)(ws + take((size_t)N * 4));
    unsigned long long* segsum = (unsigned long long*)(ws + take((size_t)N * 8));
    unsigned*           keysA  = (unsigned*)(ws + take((size_t)E * 4));
    unsigned*           valsA  = (unsigned*)(ws + take((size_t)E * 4));
    unsigned*           keysB  = (unsigned*)(ws + take((size_t)E * 4));
    unsigned*           valsB  = (unsigned*)(ws + take((size_t)E * 4));
    const int NBLK = (E + RADIX_ITEMS - 1) / RADIX_ITEMS;
    unsigned*           hist   = (unsigned*)(ws + take((size_t)256 * NBLK * 4));
    float*              clsc   = (float*)(ws + take((size_t)N * 4));

    const int TB = 256;
    const int gE  = (E + TB - 1) / TB;
    const int gN  = (N + TB - 1) / TB;
    const int g2E = (2 * E + TB - 1) / TB;

    // 1. WMMA mat-vec: both score projections in one pass over x
    matvec_wmma_kernel<<<(N + 15) / 16, 32, 0, stream>>>(x, W, s_src, s_dst, N, C);

    // 2. init segment state + cluster scores; zero output regions we accumulate into
    init_nodes_kernel<<<gN, TB, 0, stream>>>(segmax, segsum, clsc, N);
    zero_f32_kernel<<<4096, TB, 0, stream>>>(newx, (long long)N * C);
    zero_f32_kernel<<<gN, TB, 0, stream>>>((float*)new_b, (long long)N);

    // 3. segment softmax over dst (order-invariant max, fixed-point sum)
    raw_max_kernel<<<gE, TB, 0, stream>>>(src, dst, s_src, s_dst, bias, score, segmax, E);
    exp_sum_kernel<<<gE, TB, 0, stream>>>(dst, score, segmax, segsum, E);
    finalize_keys_kernel<<<gE, TB, 0, stream>>>(dst, score, segsum, keysA, valsA, E);

    // 4. stable LSD radix sort, descending score (4 x 8-bit passes, ping-pong)
    unsigned *ki = keysA, *vi = valsA, *ko = keysB, *vo = valsB;
    for (int pass = 0; pass < 4; ++pass) {
        int shift = pass * 8;
        radix_hist_kernel<<<NBLK, TB, 0, stream>>>(ki, hist, E, shift, NBLK);
        radix_scan_kernel<<<1, 32, 0, stream>>>(hist, 256 * NBLK);
        radix_scatter_kernel<<<NBLK, TB, 0, stream>>>(ki, vi, ko, vo, hist, E, shift, NBLK);
        unsigned* tk = ki; ki = ko; ko = tk;
        unsigned* tv = vi; vi = vo; vo = tv;
    }
    // result now in (ki, vi) == (keysA, valsA) after 4 swaps

    // 5. sequential greedy contraction (mask in LDS, cooperative prefetch)
    greedy_kernel<<<1, TB, 0, stream>>>(vi, src, dst, score, cluster, clsc, N, E);

    // 6. cluster aggregation
    {
        long long total = (long long)N * C;
        int g = (int)((total + TB - 1) / TB);
        newx_scatter_kernel<<<g, TB, 0, stream>>>(x, cluster, clsc, newx, N, C);
    }
    remap_edges_kernel<<<g2E, TB, 0, stream>>>(ei, cluster, new_ei, 2 * E);
    batch_scatter_kernel<<<gN, TB, 0, stream>>>(batch, cluster, new_b, N);
}